// ogb_GCN_64974265254095
// MI455X (gfx1250) — compile-verified
//
#include <hip/hip_runtime.h>
#include <hip/hip_bf16.h>

typedef float v2f __attribute__((ext_vector_type(2)));
typedef float v8f __attribute__((ext_vector_type(8)));
typedef unsigned int v4u __attribute__((ext_vector_type(4)));
typedef int v4i __attribute__((ext_vector_type(4)));
typedef int v8i __attribute__((ext_vector_type(8)));

#define N_NODES 100000
#define N_EDGES 1600000

#ifndef __has_builtin
#define __has_builtin(x) 0
#endif

// ---------------------------------------------------------------- utilities
__global__ void k_fill_zero(float* __restrict__ p, long long n) {
  long long i = (long long)blockIdx.x * blockDim.x + threadIdx.x;
  if (i < n) p[i] = 0.0f;
}

__global__ void k_degrees(const int* __restrict__ src, const int* __restrict__ dst,
                          float* __restrict__ deg_s, float* __restrict__ deg_d, int e) {
  int t = blockIdx.x * blockDim.x + threadIdx.x;
  if (t >= e) return;
  atomicAdd(&deg_s[src[t]], 1.0f);
  atomicAdd(&deg_d[dst[t]], 1.0f);
}

__global__ void k_norms(float* __restrict__ ns, float* __restrict__ nd, int n) {
  int t = blockIdx.x * blockDim.x + threadIdx.x;
  if (t >= n) return;
  float a = ns[t]; ns[t] = rsqrtf(a < 1.0f ? 1.0f : a);
  float b = nd[t]; nd[t] = rsqrtf(b < 1.0f ? 1.0f : b);
}

// ------------------------------------------------- WMMA GEMM:  Y = (X*ns) @ W
// Block = 256 threads = 8 waves; each block computes 64 output rows.
// TDM tensor_load_to_lds stages the 64x128 X tile (32 KB) and the whole
// 128xDOUT W matrix (64/20 KB) into LDS; inner loop is ds_load + v_wmma only.
// Wave w owns column tile n0 = 16*w and keeps 4 row-tile accumulators, so
// every B fragment read from LDS feeds 4 WMMAs.
template <int DOUT>
__global__ __launch_bounds__(256) void k_gemm_ns_wmma(
    const float* __restrict__ X, const float* __restrict__ ns,
    const float* __restrict__ Wm, float* __restrict__ Y, int nrows) {
  constexpr int K    = 128;
  constexpr int RT   = 4;               // row tiles per block
  constexpr int ROWS = 16 * RT;         // 64
  constexpr int TN   = (DOUT + 15) / 16;

  __shared__ float xs[ROWS * K];        // 32 KB
  __shared__ float wls[K * DOUT];       // 64 KB (128) / 20 KB (40)

  const int m0    = blockIdx.x * ROWS;
  const int trows = (nrows - m0) < ROWS ? (nrows - m0) : ROWS;

#if __has_builtin(__builtin_amdgcn_tensor_load_to_lds) && __has_builtin(__builtin_amdgcn_s_wait_tensorcnt)
  if (threadIdx.x < 32) {               // one wave issues both TDM ops (EXEC ignored)
    const v4i zero4 = {0, 0, 0, 0};
    const v8i zero8 = {0, 0, 0, 0, 0, 0, 0, 0};
    // ---- X tile: 2D tensor [nrows x 128] f32, tile [trows x 128] ----
    {
      unsigned long long ga = (unsigned long long)(uintptr_t)X +
                              (unsigned long long)m0 * K * 4ull;
      v4u g0;                                          // D# group 0
      g0.x = 1u;                                       // count=1
      g0.y = (unsigned int)(uintptr_t)xs;              // lds_addr (low 32 = LDS offset)
      g0.z = (unsigned int)ga;                         // global_addr[31:0]
      g0.w = (unsigned int)(ga >> 32) | (2u << 30);    // global_addr[56:32] | type=2
      v8i g1;                                          // D# group 1
      g1[0] = (int)(2u << 16);                         // data_size=4B, no flags
      g1[1] = (int)((unsigned)(K & 0xffff) << 16);     // tensor_dim0 lo
      g1[2] = (int)(((unsigned)K >> 16) | ((unsigned)(nrows & 0xffff) << 16)); // dim0 hi | dim1 lo
      g1[3] = (int)(((unsigned)nrows >> 16) | ((unsigned)K << 16));            // dim1 hi | tile_dim0
      g1[4] = trows;                                   // tile_dim1 | tile_dim2=0
      g1[5] = K;                                       // tensor_dim0_stride lo
      g1[6] = 0;                                       // stride0 hi | stride1 lo
      g1[7] = 0;
      __builtin_amdgcn_tensor_load_to_lds(g0, g1, zero4, zero4, zero8, 0);
    }
    // ---- W: 2D tensor [128 x DOUT] f32, tile = whole matrix ----
    {
      unsigned long long ga = (unsigned long long)(uintptr_t)Wm;
      v4u g0;
      g0.x = 1u;
      g0.y = (unsigned int)(uintptr_t)wls;
      g0.z = (unsigned int)ga;
      g0.w = (unsigned int)(ga >> 32) | (2u << 30);
      v8i g1;
      g1[0] = (int)(2u << 16);
      g1[1] = (int)((unsigned)(DOUT & 0xffff) << 16);  // tensor_dim0 = DOUT
      g1[2] = (int)((unsigned)(K & 0xffff) << 16);     // tensor_dim1 = 128
      g1[3] = (int)((unsigned)DOUT << 16);             // tile_dim0 = DOUT
      g1[4] = K;                                       // tile_dim1 = 128
      g1[5] = DOUT;                                    // tensor_dim0_stride = DOUT
      g1[6] = 0;
      g1[7] = 0;
      __builtin_amdgcn_tensor_load_to_lds(g0, g1, zero4, zero4, zero8, 0);
    }
    __builtin_amdgcn_s_wait_tensorcnt((short)0);
  }
#else
  for (int i = threadIdx.x; i < ROWS * K; i += 256) {
    int r = m0 + (i >> 7);
    xs[i] = (r < nrows) ? X[(size_t)r * K + (i & 127)] : 0.0f;
  }
  for (int i = threadIdx.x; i < K * DOUT; i += 256) wls[i] = Wm[i];
#endif
  __syncthreads();

  const int wave = threadIdx.x >> 5;
  if (wave >= TN) return;               // wave-uniform

  const int lane = threadIdx.x & 31;
  const int half = lane >> 4;           // K sub-pair select (ISA 7.12.2 f32 A/B layout)
  const int l    = lane & 15;
  const int n0   = wave << 4;
  const int col  = n0 + l;
  const int colc  = (col < DOUT) ? col : (DOUT - 1);
  const float bm  = (col < DOUT) ? 1.0f : 0.0f;

  float nsv[RT];
#pragma unroll
  for (int rt = 0; rt < RT; ++rt) {
    int r = m0 + rt * 16 + l;
    nsv[rt] = ns[r < nrows ? r : (nrows - 1)];
  }

  v8f acc[RT] = {};
#pragma unroll
  for (int k0 = 0; k0 < K; k0 += 4) {
    const int ka = k0 + (half << 1);
    v2f b;
    b.x = wls[ka * DOUT + colc] * bm;
    b.y = wls[(ka + 1) * DOUT + colc] * bm;
#pragma unroll
    for (int rt = 0; rt < RT; ++rt) {
      const float* xr = &xs[(rt * 16 + l) * K + ka];
      v2f a;
      a.x = xr[0] * nsv[rt];
      a.y = xr[1] * nsv[rt];
      acc[rt] = __builtin_amdgcn_wmma_f32_16x16x4_f32(
          false, a, false, b, (short)0, acc[rt], false, false);
    }
  }

  if (col < DOUT) {
#pragma unroll
    for (int rt = 0; rt < RT; ++rt) {
      const int rbase = m0 + rt * 16 + (half << 3);
#pragma unroll
      for (int rr = 0; rr < 8; ++rr) {
        int row = rbase + rr;
        if (row < nrows) Y[(size_t)row * DOUT + col] = acc[rt][rr];
      }
    }
  }
}

// ------------------------------------------- edge gather + scatter-add (L2 atomics)
template <int D>
__global__ void k_edge_scatter(const int* __restrict__ src, const int* __restrict__ dst,
                               const float* __restrict__ h, float* __restrict__ agg, int e) {
  const int DV = D / 4;
  long long tid = (long long)blockIdx.x * blockDim.x + threadIdx.x;
  long long total = (long long)e * DV;
  if (tid >= total) return;
  int ed = (int)(tid / DV);
  int c  = ((int)(tid - (long long)ed * DV)) << 2;
  int s = src[ed], t = dst[ed];
  const float4 v = *(const float4*)(h + (size_t)s * D + c);
  float* p = agg + (size_t)t * D + c;
  atomicAdd(p + 0, v.x);
  atomicAdd(p + 1, v.y);
  atomicAdd(p + 2, v.z);
  atomicAdd(p + 3, v.w);
}

// ---------------- x = agg * nd (in place) + per-channel sum / sumsq (D == 128)
__global__ void k_ndscale_reduce(float* __restrict__ x, const float* __restrict__ nd,
                                 float* __restrict__ sums, int nrows) {
  __shared__ float ls[256], lq[256];
  long long total  = (long long)nrows << 7;
  long long stride = (long long)gridDim.x * blockDim.x;  // multiple of 128
  float s = 0.0f, q = 0.0f;
  for (long long i = (long long)blockIdx.x * blockDim.x + threadIdx.x; i < total; i += stride) {
    int row = (int)(i >> 7);
    float v = x[i] * nd[row];
    x[i] = v;
    s += v;
    q += v * v;
  }
  ls[threadIdx.x] = s;
  lq[threadIdx.x] = q;
  __syncthreads();
  if (threadIdx.x < 128) {
    atomicAdd(&sums[threadIdx.x],       ls[threadIdx.x] + ls[threadIdx.x + 128]);
    atomicAdd(&sums[128 + threadIdx.x], lq[threadIdx.x] + lq[threadIdx.x + 128]);
  }
}

__global__ void k_bn_params(const float* __restrict__ sums, const float* __restrict__ g,
                            const float* __restrict__ beta, float* __restrict__ ab, int nrows) {
  int c = threadIdx.x;
  float inv_n = 1.0f / (float)nrows;
  float mean = sums[c] * inv_n;
  float var  = sums[128 + c] * inv_n - mean * mean;
  float is   = rsqrtf(var + 1e-5f);
  float A    = g[c] * is;
  ab[c]       = A;
  ab[128 + c] = beta[c] - mean * A;
}

__global__ void k_bn_relu(float* __restrict__ x, const float* __restrict__ ab, long long total) {
  long long i = (long long)blockIdx.x * blockDim.x + threadIdx.x;
  if (i >= total) return;
  int c = (int)(i & 127);
  float v = x[i] * ab[c] + ab[128 + c];
  x[i] = v > 0.0f ? v : 0.0f;
}

__global__ void k_final(const float* __restrict__ agg, const float* __restrict__ nd,
                        const float* __restrict__ b2, float* __restrict__ out, int nrows) {
  const int D = 40;
  long long i = (long long)blockIdx.x * blockDim.x + threadIdx.x;
  long long total = (long long)nrows * D;
  if (i >= total) return;
  int row = (int)(i / D);
  int c   = (int)(i - (long long)row * D);
  out[i] = agg[i] * nd[row] + b2[c];
}

// ---------------------------------------------------------------- launcher
extern "C" void kernel_launch(void* const* d_in, const int* in_sizes, int n_in,
                              void* d_out, int out_size, void* d_ws, size_t ws_size,
                              hipStream_t stream) {
  (void)in_sizes; (void)n_in; (void)out_size; (void)ws_size;
  const float* feat = (const float*)d_in[0];
  const int*   esrc = (const int*)d_in[1];
  const int*   edst = (const int*)d_in[2];
  const float* W0   = (const float*)d_in[3];
  const float* W1   = (const float*)d_in[4];
  const float* W2   = (const float*)d_in[5];
  const float* b2   = (const float*)d_in[6];
  const float* g0   = (const float*)d_in[7];
  const float* be0  = (const float*)d_in[8];
  const float* g1   = (const float*)d_in[9];
  const float* be1  = (const float*)d_in[10];
  float* out = (float*)d_out;

  const int n = N_NODES, e = N_EDGES;
  float* ws   = (float*)d_ws;
  float* ns   = ws;                                  // n
  float* nd   = ns + n;                              // n
  float* bufA = nd + n;                              // n*128
  float* bufB = bufA + (size_t)n * 128;              // n*128
  float* sums = bufB + (size_t)n * 128;              // 256
  float* ab   = sums + 256;                          // 256

  auto cdiv = [](long long a, long long b) -> int { return (int)((a + b - 1) / b); };
  const int BT = 256;

  // degree norms
  k_fill_zero<<<cdiv(2LL * n, BT), BT, 0, stream>>>(ns, 2LL * n);
  k_degrees  <<<cdiv(e, BT),      BT, 0, stream>>>(esrc, edst, ns, nd, e);
  k_norms    <<<cdiv(n, BT),      BT, 0, stream>>>(ns, nd, n);

  const long long hid = (long long)n * 128;
  const int gemm_blocks = cdiv(n, 64);

  // ---------- layer 0: feat -> bufB -> scatter -> bufA ----------
  k_gemm_ns_wmma<128><<<gemm_blocks, BT, 0, stream>>>(feat, ns, W0, bufB, n);
  k_fill_zero<<<cdiv(hid, BT), BT, 0, stream>>>(bufA, hid);
  k_edge_scatter<128><<<cdiv((long long)e * 32, BT), BT, 0, stream>>>(esrc, edst, bufB, bufA, e);
  k_fill_zero<<<1, BT, 0, stream>>>(sums, 256);
  k_ndscale_reduce<<<1024, BT, 0, stream>>>(bufA, nd, sums, n);
  k_bn_params<<<1, 128, 0, stream>>>(sums, g0, be0, ab, n);
  k_bn_relu<<<cdiv(hid, BT), BT, 0, stream>>>(bufA, ab, hid);

  // ---------- layer 1: bufA -> bufB -> scatter -> bufA ----------
  k_gemm_ns_wmma<128><<<gemm_blocks, BT, 0, stream>>>(bufA, ns, W1, bufB, n);
  k_fill_zero<<<cdiv(hid, BT), BT, 0, stream>>>(bufA, hid);
  k_edge_scatter<128><<<cdiv((long long)e * 32, BT), BT, 0, stream>>>(esrc, edst, bufB, bufA, e);
  k_fill_zero<<<1, BT, 0, stream>>>(sums, 256);
  k_ndscale_reduce<<<1024, BT, 0, stream>>>(bufA, nd, sums, n);
  k_bn_params<<<1, 128, 0, stream>>>(sums, g1, be1, ab, n);
  k_bn_relu<<<cdiv(hid, BT), BT, 0, stream>>>(bufA, ab, hid);

  // ---------- layer 2: bufA -> bufB (D=40) -> scatter -> bufA -> out ----------
  const long long outelts = (long long)n * 40;
  k_gemm_ns_wmma<40><<<gemm_blocks, BT, 0, stream>>>(bufA, ns, W2, bufB, n);
  k_fill_zero<<<cdiv(outelts, BT), BT, 0, stream>>>(bufA, outelts);
  k_edge_scatter<40><<<cdiv((long long)e * 10, BT), BT, 0, stream>>>(esrc, edst, bufB, bufA, e);
  k_final<<<cdiv(outelts, BT), BT, 0, stream>>>(bufA, nd, b2, out, n);
}